// AdaptiveComputationTime_78185584656712
// MI455X (gfx1250) — compile-verified
//
#include <hip/hip_runtime.h>
#include <hip/hip_bf16.h>
#include <math.h>
#include <stdint.h>

#define ACT_B 8
#define ACT_T 12
#define ACT_S 4096
#define ACT_D 512
#define ACT_NROW (ACT_B * ACT_S)   // 32768
#define ACT_THRESH 0.99f

// ---------------- CDNA5 async global->LDS copy helpers ----------------
typedef int v4i __attribute__((vector_size(16)));
typedef float f32x4 __attribute__((ext_vector_type(4)));
typedef __attribute__((address_space(1))) v4i* gv4p;   // global ptr to int4
typedef __attribute__((address_space(3))) v4i* lv4p;   // LDS ptr to int4

#if __has_builtin(__builtin_amdgcn_global_load_async_to_lds_b128)
#define ASYNC_B128(gp, lp, OFF)                                              \
  __builtin_amdgcn_global_load_async_to_lds_b128(                            \
      (gv4p)(uintptr_t)(gp),                                                 \
      (lv4p)(uint32_t)(uintptr_t)(lp),                                       \
      (OFF), 0)
#else
#define ASYNC_B128(gp, lp, OFF)                                              \
  asm volatile("global_load_async_to_lds_b128 %0, %1, off offset:" #OFF      \
               :: "v"((uint32_t)(uintptr_t)(lp)), "v"((void*)(gp))           \
               : "memory")
#endif

#if __has_builtin(__builtin_amdgcn_s_wait_asynccnt)
#define WAIT_ASYNC(N)                                                        \
  do {                                                                       \
    asm volatile("" ::: "memory");                                           \
    __builtin_amdgcn_s_wait_asynccnt(N);                                     \
    asm volatile("" ::: "memory");                                           \
  } while (0)
#else
#define WAIT_ASYNC(N) asm volatile("s_wait_asynccnt %0" :: "i"(N) : "memory")
#endif

#define WAIT_DSCNT0() asm volatile("s_wait_dscnt 0" ::: "memory")

// Issue one 2KB row (512 floats) as 4 async b128 ops: lane l moves 16B at
// byte offset l*16 + i*512; INST_OFFSET applies to both global and LDS addr.
__device__ __forceinline__ void issue_row(const float* g, float* l, int lane) {
  const float* gp = g + lane * 4;
  float* lp = l + lane * 4;
  ASYNC_B128(gp, lp, 0);
  ASYNC_B128(gp, lp, 512);
  ASYNC_B128(gp, lp, 1024);
  ASYNC_B128(gp, lp, 1536);
}

// ---------------- Kernel 1: fused ACT scan, one wave per (b,s) row ----------
__global__ __launch_bounds__(256) void act_scan_kernel(
    const float* __restrict__ hs,   // (B,T,S,D)
    const float* __restrict__ Wh,   // (D,)
    const float* __restrict__ bh,   // (1,)
    float* __restrict__ out,        // (B,S,D)
    float* __restrict__ st,         // [3*NROW]: cum | n_up | running
    unsigned int* __restrict__ flag) {
  __shared__ __align__(16) float sh[8 * 2 * ACT_D];  // 8 waves * 2 bufs * 512f = 32KB

  const int lane = threadIdx.x & 31;
  const int wv   = threadIdx.x >> 5;
  const int row  = blockIdx.x * 8 + wv;          // 0..32767  (= b*S + s)
  const int b    = row >> 12;
  const int s    = row & (ACT_S - 1);

  float* buf = &sh[wv * (2 * ACT_D)];

  // Per-lane weight slice: d = lane*4 + j*128
  float4 w[4];
#pragma unroll
  for (int j = 0; j < 4; ++j)
    w[j] = *(const float4*)&Wh[lane * 4 + j * 128];
  const float bias = bh[0];

  const size_t tstride = (size_t)ACT_S * ACT_D;  // 2M floats
  const float* g0 = hs + (size_t)b * ACT_T * tstride + (size_t)s * ACT_D;

  // Prime double buffer: t=0 -> buf0, t=1 -> buf1
  issue_row(g0, buf, lane);
  issue_row(g0 + tstride, buf + ACT_D, lane);

  float4 acc[4];
#pragma unroll
  for (int j = 0; j < 4; ++j) acc[j] = make_float4(0.f, 0.f, 0.f, 0.f);

  float cum = 0.f, nup = 0.f, runn = 1.f;

#pragma unroll
  for (int t = 0; t < ACT_T; ++t) {
    // In-order async completion: <=4 outstanding means batch t has landed.
    if (t < ACT_T - 1) { WAIT_ASYNC(4); } else { WAIT_ASYNC(0); }

    const float* cb = buf + (t & 1) * ACT_D;
    float4 v[4];
#pragma unroll
    for (int j = 0; j < 4; ++j)
      v[j] = *(const float4*)&cb[lane * 4 + j * 128];

    // Ensure our ds reads finished before async DMA re-targets this buffer.
    WAIT_DSCNT0();
    if (t + 2 < ACT_T)
      issue_row(g0 + (size_t)(t + 2) * tstride, buf + (t & 1) * ACT_D, lane);

    float part = 0.f;
#pragma unroll
    for (int j = 0; j < 4; ++j) {
      part = fmaf(v[j].x, w[j].x, part);
      part = fmaf(v[j].y, w[j].y, part);
      part = fmaf(v[j].z, w[j].z, part);
      part = fmaf(v[j].w, w[j].w, part);
    }
    // wave32 all-reduce
#pragma unroll
    for (int off = 16; off >= 1; off >>= 1)
      part += __shfl_xor(part, off, 32);

    float p = 1.f / (1.f + __expf(-(part + bias)));
    p *= runn;
    const float ncum = cum + p;
    const bool halt = (ncum >= ACT_THRESH) && (runn > 0.f);
    const float rem = halt ? (1.f - cum) : p;

#pragma unroll
    for (int j = 0; j < 4; ++j) {
      acc[j].x = fmaf(rem, v[j].x, acc[j].x);
      acc[j].y = fmaf(rem, v[j].y, acc[j].y);
      acc[j].z = fmaf(rem, v[j].z, acc[j].z);
      acc[j].w = fmaf(rem, v[j].w, acc[j].w);
    }
    nup += runn;
    if (halt) runn = 0.f;
    cum = ncum;
  }

  // Streaming output: written once, not re-read in the expected path -> NT.
  float* orow = out + (size_t)row * ACT_D;
#pragma unroll
  for (int j = 0; j < 4; ++j) {
    f32x4 tmp = {acc[j].x, acc[j].y, acc[j].z, acc[j].w};
    __builtin_nontemporal_store(tmp, (f32x4*)&orow[lane * 4 + j * 128]);
  }

  if (lane == 0) {
    st[row] = cum;
    st[ACT_NROW + row] = nup;
    st[2 * ACT_NROW + row] = runn;
    if (runn > 0.f) atomicOr(flag, 1u);
  }
}

// ---------------- Kernel 2: global any(running>0) branch + effective_steps --
__global__ __launch_bounds__(256) void act_finalize_kernel(
    const float* __restrict__ hs,
    const float* __restrict__ st,
    const unsigned int* __restrict__ flag,
    float* __restrict__ out,
    float* __restrict__ eff) {
  const int lane = threadIdx.x & 31;
  const int wv   = threadIdx.x >> 5;
  const int row  = blockIdx.x * 8 + wv;

  const float cum  = st[row];
  float nup        = st[ACT_NROW + row];
  const float runn = st[2 * ACT_NROW + row];

  if (*flag != 0u) {
    // out += (1-cum) * h_last for ALL rows; n_up += running
    const size_t tstride = (size_t)ACT_S * ACT_D;
    const int b = row >> 12;
    const int s = row & (ACT_S - 1);
    const float* hl =
        hs + ((size_t)b * ACT_T + (ACT_T - 1)) * tstride + (size_t)s * ACT_D;
    float* orow = out + (size_t)row * ACT_D;
    const float c = 1.f - cum;
#pragma unroll
    for (int j = 0; j < 4; ++j) {
      const int d = lane * 4 + j * 128;
      float4 h = *(const float4*)&hl[d];
      float4 o = *(float4*)&orow[d];
      o.x = fmaf(c, h.x, o.x);
      o.y = fmaf(c, h.y, o.y);
      o.z = fmaf(c, h.z, o.z);
      o.w = fmaf(c, h.w, o.w);
      *(float4*)&orow[d] = o;
    }
    nup += runn;
  }
  if (lane == 0)
    eff[row] = nup + fmaxf(1.f - cum, 0.f);
}

// ---------------- Kernel 3: deterministic ponder_cost = mean_s(eff) ---------
__global__ __launch_bounds__(256) void ponder_kernel(
    const float* __restrict__ eff, float* __restrict__ pc) {
  __shared__ float red[256];
  const int b = blockIdx.x;
  const int tid = threadIdx.x;
  float sum = 0.f;
  for (int k = tid; k < ACT_S; k += 256)
    sum += eff[b * ACT_S + k];
  red[tid] = sum;
  __syncthreads();
  for (int off = 128; off > 0; off >>= 1) {
    if (tid < off) red[tid] += red[tid + off];
    __syncthreads();
  }
  if (tid == 0) pc[b] = red[0] * (1.f / ACT_S);
}

// ---------------- Launch ----------------------------------------------------
extern "C" void kernel_launch(void* const* d_in, const int* in_sizes, int n_in,
                              void* d_out, int out_size, void* d_ws,
                              size_t ws_size, hipStream_t stream) {
  (void)in_sizes; (void)n_in; (void)out_size; (void)ws_size;
  const float* hs = (const float*)d_in[0];
  const float* Wh = (const float*)d_in[1];
  const float* bh = (const float*)d_in[2];

  float* out = (float*)d_out;                         // B*S*D
  float* pc  = out + (size_t)ACT_NROW * ACT_D;        // B
  float* eff = pc + ACT_B;                            // B*S

  float* st = (float*)d_ws;                           // 3 * NROW floats
  unsigned int* flag = (unsigned int*)(st + 3 * ACT_NROW);

  (void)hipMemsetAsync(flag, 0, sizeof(unsigned int), stream);
  act_scan_kernel<<<ACT_NROW / 8, 256, 0, stream>>>(hs, Wh, bh, out, st, flag);
  act_finalize_kernel<<<ACT_NROW / 8, 256, 0, stream>>>(hs, st, flag, out, eff);
  ponder_kernel<<<ACT_B, 256, 0, stream>>>(eff, pc);
}